// TopologicalRegularizer_6270652252581
// MI455X (gfx1250) — compile-verified
//
#include <hip/hip_runtime.h>
#include <stdint.h>

#define B_IMG 4
#define H_IMG 256
#define W_IMG 256
#define N_PIX (H_IMG*W_IMG)   /* 65536 */
#define TOPK 12
#define TARGET_LT 0.5f

typedef __attribute__((ext_vector_type(2))) float v2f;
typedef __attribute__((ext_vector_type(8))) float v8f;

// ---------------------------------------------------------------------------
// Kernel A: filt = 1 - prob*roi ; key = (floatbits << 16) | pixel_index
// Non-negative float bit patterns are order-isomorphic to values, and the
// embedded index makes a plain ascending sort equivalent to a stable argsort.
// ---------------------------------------------------------------------------
__global__ void ka_build_keys(const float* __restrict__ prob,
                              const float* __restrict__ roi,
                              unsigned long long* __restrict__ keys)
{
    int t = blockIdx.x * blockDim.x + threadIdx.x;
    if (t >= B_IMG * N_PIX) return;
    float f = 1.0f - prob[t] * roi[t];
    unsigned int bits = __float_as_uint(f);
    keys[t] = ((unsigned long long)bits << 16) | (unsigned long long)(t & (N_PIX - 1));
}

// ---------------------------------------------------------------------------
// Kernel B: per-image stable LSD radix sort of 48-bit keys, 12 x 4-bit passes.
// 1024 threads (32 waves) per image; each thread owns 64 contiguous elements.
// Cross-wave histogram scan done hierarchically; the heavy level (exclusive
// offsets of 4-wave groups + per-bin grand totals) is computed on the matrix
// pipe with chained V_WMMA_F32_16X16X4_F32, A = all-ones (layout-robust
// column-sum reduction: D[m][n] = sum_k B[k][n] + C[m][n]).
// ---------------------------------------------------------------------------
__global__ __launch_bounds__(1024)
void kb_radix_sort(unsigned long long* __restrict__ keysA,
                   unsigned long long* __restrict__ keysB)
{
    const int img = blockIdx.x;
    unsigned long long* bufA = keysA + (size_t)img * N_PIX;
    unsigned long long* bufB = keysB + (size_t)img * N_PIX;
    const int t    = threadIdx.x;
    const int lane = t & 31;
    const int w    = t >> 5;              // wave id 0..31

    __shared__ int            cw[32][16];       // per-wave digit counts
    __shared__ int            gOff[9][16];      // exclusive 4-wave-group offsets, [8]=totals
    __shared__ int            digitBase[16];    // exclusive scan of totals over bins
    __shared__ unsigned short thOff[1024 * 16]; // per-thread digit counters / offsets

    for (int pass = 0; pass < 12; ++pass) {
        unsigned long long* src = (pass & 1) ? bufB : bufA;
        unsigned long long* dst = (pass & 1) ? bufA : bufB;
        const int shift = pass * 4;
        const int base  = t * 64;

        // ---- per-thread histogram (in LDS: dynamic digit index) ----
        #pragma unroll
        for (int d = 0; d < 16; ++d) thOff[t * 16 + d] = 0;
        for (int j = 0; j < 64; ++j) {
            int d = (int)((src[base + j] >> shift) & 15ull);
            thOff[t * 16 + d]++;
        }

        // ---- per-bin exclusive scan across lanes within each wave ----
        #pragma unroll
        for (int d = 0; d < 16; ++d) {
            int inc  = (int)thOff[t * 16 + d];
            int scan = inc;
            #pragma unroll
            for (int off = 1; off < 32; off <<= 1) {
                int u = __shfl_up(scan, off, 32);
                if (lane >= off) scan += u;
            }
            int tot = __shfl(scan, 31, 32);
            if (lane == d) cw[w][d] = tot;
            thOff[t * 16 + d] = (unsigned short)(scan - inc);  // lane-exclusive
        }
        __syncthreads();

        // ---- wave 0: hierarchical cross-wave scan, heavy level on WMMA ----
        if (t < 32) {
            const int b = lane & 15;
            const int h = lane >> 4;     // which of two lane-halves
#if defined(__gfx1250__) && __has_builtin(__builtin_amdgcn_wmma_f32_16x16x4_f32)
            v8f acc = {};
            v2f Aones; Aones[0] = 1.0f; Aones[1] = 1.0f;
            #pragma unroll
            for (int k = 0; k < 8; ++k) {
                if (lane < 16) gOff[k][lane] = (int)acc[0];   // exclusive group offset
                v2f Bv;
                Bv[0] = (float)cw[4 * k + h][b];
                Bv[1] = (float)cw[4 * k + 2 + h][b];
                // D[m][n] = sum_k B[k][n] + C : per-bin sum of 4 waves on matrix pipe
                acc = __builtin_amdgcn_wmma_f32_16x16x4_f32(
                          false, Aones, false, Bv, (short)0, acc, false, false);
            }
            float totf = acc[0];
#else
            float totf = 0.0f;
            #pragma unroll
            for (int k = 0; k < 8; ++k) {
                if (lane < 16) gOff[k][lane] = (int)totf;
                float v = (float)cw[4 * k + h][b] + (float)cw[4 * k + 2 + h][b];
                v += __shfl_xor(v, 16, 32);
                totf += v;
            }
#endif
            if (lane < 16) gOff[8][lane] = (int)totf;
            // exclusive scan of per-bin totals over 16 bins (lanes 0..15)
            float sv = (lane < 16) ? totf : 0.0f;
            float run = sv;
            #pragma unroll
            for (int off = 1; off < 16; off <<= 1) {
                float u = __shfl_up(run, off, 32);
                if (lane >= off) run += u;
            }
            if (lane < 16) digitBase[lane] = (int)(run - sv);
        }
        __syncthreads();

        // ---- finalize per-thread scatter offsets, then stable scatter ----
        {
            const int g  = w >> 2;
            const int gs = g << 2;
            #pragma unroll
            for (int d = 0; d < 16; ++d) {
                int pre = 0;
                for (int i = gs; i < w; ++i) pre += cw[i][d];   // <=3 waves in group
                thOff[t * 16 + d] = (unsigned short)((int)thOff[t * 16 + d] +
                                     digitBase[d] + gOff[g][d] + pre);
            }
            for (int j = 0; j < 64; ++j) {
                unsigned long long k = src[base + j];
                int d   = (int)((k >> shift) & 15ull);
                int pos = (int)thOff[t * 16 + d];
                thOff[t * 16 + d] = (unsigned short)(pos + 1);
                dst[pos] = k;
            }
        }
        __syncthreads();
    }
}

// ---------------------------------------------------------------------------
// Kernel C: per-image serial union-find (elder rule) over sorted pixels.
// parent[] lives in 256KB of dynamic LDS (CDNA5 WGP has 320KB) so the hot
// pointer-chase never leaves the WGP. Sorted keys are streamed with
// global_prefetch ahead of the scalar consumer. Top-12 lifetimes maintained
// inline; pair multiset at a death pixel is neighbor-order invariant under
// the elder rule, so this matches the reference exactly.
// ---------------------------------------------------------------------------
__global__ __launch_bounds__(1024)
void kc_union_find(const unsigned long long* __restrict__ keys,
                   int* __restrict__ birthrank,
                   float* __restrict__ perImg)
{
    extern __shared__ int parent[];    // N_PIX ints = 256 KB dynamic LDS
    const int img = blockIdx.x;
    const unsigned long long* k = keys + (size_t)img * N_PIX;
    int* br = birthrank + (size_t)img * N_PIX;

    for (int i = threadIdx.x; i < N_PIX; i += blockDim.x) parent[i] = -1;
    __syncthreads();
    if (threadIdx.x != 0) return;

    float top[TOPK];
    #pragma unroll
    for (int i = 0; i < TOPK; ++i) top[i] = -3.0e38f;
    int cnt = 0;

    for (int s = 0; s < N_PIX; ++s) {
        __builtin_prefetch(k + ((s + 96) & (N_PIX - 1)), 0, 1);  // global_prefetch_b8
        unsigned long long key = k[s];
        int   p  = (int)(key & 0xFFFFull);
        float fp = __uint_as_float((unsigned int)(key >> 16));
        parent[p] = p;
        br[p]     = s;
        int r = p >> 8, cc = p & (W_IMG - 1);
        int nq[4];
        nq[0] = (r > 0)         ? p - W_IMG : -1;
        nq[1] = (r < H_IMG - 1) ? p + W_IMG : -1;
        nq[2] = (cc > 0)        ? p - 1     : -1;
        nq[3] = (cc < W_IMG - 1)? p + 1     : -1;
        #pragma unroll
        for (int e = 0; e < 4; ++e) {
            int q = nq[e];
            if (q < 0) continue;
            if (parent[q] < 0) continue;
            int rp = p;
            while (parent[rp] != rp) { int g2 = parent[parent[rp]]; parent[rp] = g2; rp = g2; }
            int rq = q;
            while (parent[rq] != rq) { int g2 = parent[parent[rq]]; parent[rq] = g2; rq = g2; }
            if (rp == rq) continue;
            int brp = br[rp], brq = br[rq];
            int old_, yng, byng;
            if (brp <= brq) { old_ = rp; yng = rq; byng = brq; }
            else            { old_ = rq; yng = rp; byng = brp; }
            float fb = __uint_as_float((unsigned int)(k[byng] >> 16));
            float lt = fp - fb;
            ++cnt;
            if (lt > top[0]) {                 // keep 12 largest, sorted ascending
                top[0] = lt;
                #pragma unroll
                for (int j = 1; j < TOPK; ++j) {
                    if (top[j - 1] > top[j]) { float tm = top[j]; top[j] = top[j - 1]; top[j - 1] = tm; }
                }
            }
            parent[yng] = old_;
        }
    }

    int keff = cnt < TOPK ? cnt : TOPK;
    float per = 0.0f;
    if (cnt > 0) {
        for (int j = TOPK - keff; j < TOPK; ++j) {
            float d = top[j] - TARGET_LT;
            per += d * d;
        }
        per /= (float)keff;
    }
    perImg[img] = per;
}

// ---------------------------------------------------------------------------
// Kernel D: batch mean.
// ---------------------------------------------------------------------------
__global__ void kd_final(const float* __restrict__ perImg, float* __restrict__ out)
{
    if (threadIdx.x == 0 && blockIdx.x == 0)
        out[0] = (perImg[0] + perImg[1] + perImg[2] + perImg[3]) * 0.25f;
}

// ---------------------------------------------------------------------------
extern "C" void kernel_launch(void* const* d_in, const int* in_sizes, int n_in,
                              void* d_out, int out_size, void* d_ws, size_t ws_size,
                              hipStream_t stream)
{
    const float* prob = (const float*)d_in[0];
    const float* roi  = (const float*)d_in[1];
    char* ws = (char*)d_ws;

    // workspace layout (all L2-resident: ~5 MB total)
    unsigned long long* keysA = (unsigned long long*)(ws + 0);            // 2 MB
    unsigned long long* keysB = (unsigned long long*)(ws + (2u << 20));   // 2 MB
    int*   birthrank          = (int*)  (ws + (4u << 20));                // 1 MB
    float* perImg             = (float*)(ws + (5u << 20));                // 16 B

    ka_build_keys<<<(B_IMG * N_PIX + 255) / 256, 256, 0, stream>>>(prob, roi, keysA);
    kb_radix_sort<<<B_IMG, 1024, 0, stream>>>(keysA, keysB);
    // 12 passes (even) -> sorted result ends in keysA
    kc_union_find<<<B_IMG, 1024, N_PIX * sizeof(int), stream>>>(keysA, birthrank, perImg);
    kd_final<<<1, 32, 0, stream>>>(perImg, (float*)d_out);
}